// ConvolutionBlock_45268955300397
// MI455X (gfx1250) — compile-verified
//
#include <hip/hip_runtime.h>
#include <math.h>

// ---------------------------------------------------------------------------
// ConvolutionBlock on gfx1250: implicit-GEMM conv3d via V_WMMA_F32_16X16X4_F32
// with GLOBAL_LOAD_ASYNC_TO_LDS staging of the input halo tile.
// ---------------------------------------------------------------------------

typedef __attribute__((ext_vector_type(2))) float v2f;
typedef __attribute__((ext_vector_type(8))) float v8f;

#define S 64
#define S3 262144              // 64^3
#define COUT 48                // NS + NG + 3*NV
#define NSC 16
#define NGC 8
#define NVC 8
#define ROWLEN 66              // 64 voxels + 2 halo

__device__ __forceinline__ float eps3f(int a, int b, int c) {
  if (a == b || b == c || a == c) return 0.f;
  return (((b - a + 3) % 3) == 1) ? 1.f : -1.f;
}

// emb (4 bump-basis values) and sh1 (sqrt3 * rhat) for lattice point p
__device__ __forceinline__ void lattice_p(int p, float* embp, float* sh1p) {
  int kd = p / 9, kh = (p / 3) % 3, kw = p % 3;
  float gx = (float)(kd - 1), gy = (float)(kh - 1), gz = (float)(kw - 1);
  float r = sqrtf(gx * gx + gy * gy + gz * gz);
  float inv = (r > 0.f) ? (1.f / fmaxf(r, 1e-9f)) : 0.f;
  const float s3 = 1.7320508075688772f;
  sh1p[0] = s3 * gx * inv;
  sh1p[1] = s3 * gy * inv;
  sh1p[2] = s3 * gz * inv;
  const float step = 1.f / 3.f;
  for (int b = 0; b < 4; ++b) {
    float dc = (r - (float)b * step) / step;
    float d2 = dc * dc;
    embp[b] = (fabsf(dc) < 1.f)
                  ? 1.14136f * expf(2.f) * expf(-1.f / fmaxf(1.f - d2, 1e-9f))
                  : 0.f;
  }
}

// ---------------------------------------------------------------------------
// Prep: build Km1 (27x8x48) and Km2 (27x40x48) with skips folded into the
// center tap (p==13); also zero the 224-float stats/finalize region.
// ---------------------------------------------------------------------------
__global__ void prep_kernel(const float* __restrict__ w1, const float* __restrict__ sc1,
                            const float* __restrict__ w2, const float* __restrict__ sc2_s,
                            const float* __restrict__ sc2_v,
                            float* __restrict__ Km1, float* __restrict__ Km2,
                            float* __restrict__ statsZero) {
  const int idx0 = blockIdx.x * blockDim.x + threadIdx.x;
  const int stride = gridDim.x * blockDim.x;
  const float inv33 = 0.19245008972987526f;   // 1/3^1.5
  const float a1 = 0.35355339059327373f;      // 1/sqrt(8)
  const float as2 = 0.20412414523193154f;     // 1/sqrt(24)
  const float av2 = 0.17677669529663687f;     // 1/sqrt(32)
  const float i_s3 = 0.5773502691896258f;     // 1/sqrt(3)
  const float i_s6 = 0.4082482904638631f;     // 1/sqrt(6)

  for (int t = idx0; t < 224; t += stride) statsZero[t] = 0.f;

  // Km1: entry e = (p*8 + i)*48 + n
  for (int e = idx0; e < 27 * 8 * COUT; e += stride) {
    int n = e % COUT;
    int t = e / COUT;
    int i = t % 8;
    int p = t / 8;
    float embp[4], sh1p[3];
    lattice_p(p, embp, sh1p);
    float val;
    if (n < NSC) {
      float w = 0.f;
      for (int b = 0; b < 4; ++b) w += embp[b] * w1[b * 256 + i * 16 + n];
      val = a1 * w * inv33;
    } else if (n < NSC + NGC) {
      float w = 0.f;
      for (int b = 0; b < 4; ++b) w += embp[b] * w1[b * 256 + 128 + i * 8 + (n - 16)];
      val = a1 * w * inv33;
    } else {
      int u = (n - 24) / 3, mc = (n - 24) % 3;
      float w = 0.f;
      for (int b = 0; b < 4; ++b) w += embp[b] * w1[b * 256 + 192 + i * 8 + u];
      val = a1 * w * inv33 * sh1p[mc];
    }
    if (p == 13 && n < 24) val += sc1[i * 24 + n] * a1;   // sc1 skip folded
    Km1[e] = val;
  }

  // Km2: entry e = (p*40 + r)*48 + n  (r<16 scalar rows, r=16+u*3+mc vector)
  for (int e = idx0; e < 27 * 40 * COUT; e += stride) {
    int n = e % COUT;
    int t = e / COUT;
    int r = t % 40;
    int p = t / 40;
    float embp[4], sh1p[3];
    lattice_p(p, embp, sh1p);
    auto W2 = [&](int col) {
      float s = 0.f;
      for (int b = 0; b < 4; ++b) s += embp[b] * w2[b * 832 + col];
      return s * inv33;
    };
    float val;
    if (r < 16) {
      if (n < 16) {
        val = as2 * W2(r * 16 + n);
      } else if (n < 24) {
        val = as2 * W2(256 + r * 8 + (n - 16));
      } else {
        int u = (n - 24) / 3, nc = (n - 24) % 3;
        val = av2 * W2(384 + r * 8 + u) * sh1p[nc];
      }
      if (p == 13 && n < 24) val += sc2_s[r * 24 + n] * 0.25f;   // sc2_s skip
    } else {
      int u = (r - 16) / 3, mc = (r - 16) % 3;
      if (n < 16) {
        val = as2 * i_s3 * W2(576 + u * 16 + n) * sh1p[mc];
      } else if (n < 24) {
        val = as2 * i_s3 * W2(704 + u * 8 + (n - 16)) * sh1p[mc];
      } else {
        int j = (n - 24) / 3, nc = (n - 24) % 3;
        val = (mc == nc) ? av2 * W2(512 + u * 8 + j) : 0.f;       // kv0
        float ec = 0.f;
        for (int s = 0; s < 3; ++s) ec += eps3f(mc, s, nc) * sh1p[s];
        val += av2 * i_s6 * W2(768 + u * 8 + j) * ec;             // kv1
        if (p == 13 && mc == nc) val += sc2_v[u * 8 + j] * a1;    // sc2_v skip
      }
    }
    Km2[e] = val;
  }
}

// ---------------------------------------------------------------------------
// Implicit-GEMM 3x3x3 conv via WMMA f32 16x16x4.
// Block = 64 threads (2 waves); block covers one full (d,h) row of 64 voxels.
// Each wave computes TWO 16-voxel M-tiles (32 voxels) x 48 channels, so each
// B fragment (6 global loads / K-step) feeds 6 WMMAs.
// Input halo tile staged in LDS via GLOBAL_LOAD_ASYNC_TO_LDS_B32 (ASYNCcnt).
// K-ordering: k = p*cin + r  ->  B = Km[(p*cin+r)*48 + n] (contiguous).
// Accumulates BN statistics (sum / sumsq for scalar ch, vec sumsq per group).
// ---------------------------------------------------------------------------
__global__ void conv_wmma_kernel(const float* __restrict__ x, const float* __restrict__ Km,
                                 float* __restrict__ out, float* __restrict__ stats,
                                 int cin) {
  extern __shared__ float smem[];
  float* tile = smem;                          // cin*9*66   (LDS offset 0)
  float* sSum = smem + cin * 9 * ROWLEN;       // 48
  float* sSq = sSum + COUT;                    // 48
  int* koff = (int*)(sSq + COUT);              // cin*27

  const int bid = blockIdx.x;                  // 0..4095 = (d,h)
  const int d = bid >> 6;
  const int h = bid & 63;
  const int tid = threadIdx.x;
  const int nth = blockDim.x;
  const int KT = cin * 27;

  // per-k LDS offset table: k = p*cin + r -> row (r*9 + kd*3 + kh), + kw
  for (int k = tid; k < KT; k += nth) {
    int p = k / cin, r = k - p * cin;
    int kd = p / 9, kh = (p / 3) % 3, kw = p % 3;
    koff[k] = (r * 9 + kd * 3 + kh) * ROWLEN + kw;
  }
  for (int t = tid; t < 2 * COUT; t += nth) sSum[t] = 0.f;

  // Stage input tile with halo. In-bounds elements: async copy global->LDS
  // (no VGPR round trip, tracked by ASYNCcnt). OOB halo: zero via ds_store.
  const int totalTile = cin * 9 * ROWLEN;
  for (int t = tid; t < totalTile; t += nth) {
    int rr = t / ROWLEN;
    int wl = t - rr * ROWLEN;
    int i = rr / 9;
    int rem = rr - i * 9;
    int kd = rem / 3, kh = rem % 3;
    int dd = d + kd - 1, hh = h + kh - 1, ww = wl - 1;
    if (dd >= 0 && dd < S && hh >= 0 && hh < S && ww >= 0 && ww < S) {
      const float* gp = x + ((size_t)(i * S + dd) * S + hh) * S + ww;
      unsigned ldsOff = (unsigned)t * 4u;      // dynamic LDS starts at offset 0
      asm volatile("global_load_async_to_lds_b32 %0, %1, off"
                   :: "v"(ldsOff), "v"(gp) : "memory");
    } else {
      tile[t] = 0.f;
    }
  }
  asm volatile("s_wait_asynccnt 0x0" ::: "memory");
  __syncthreads();

  const int lane = tid & 31;
  const int wave = tid >> 5;
  const int hi = lane >> 4;     // 0: K rows k0,k0+1 ; 1: k0+2,k0+3
  const int lo = lane & 15;
  const int m0 = wave * 32;     // first M-tile voxel base
  const int m1 = m0 + 16;       // second M-tile voxel base

  v8f accA0 = {}, accA1 = {}, accA2 = {};   // M-tile 0, N-tiles 0..2
  v8f accB0 = {}, accB1 = {}, accB2 = {};   // M-tile 1, N-tiles 0..2
  for (int k0 = 0; k0 < KT; k0 += 4) {
    int ka = k0 + 2 * hi;
    int o0 = koff[ka], o1 = koff[ka + 1];
    // A 16x4 f32: lanes 0-15 hold (M=lo, K=k0 / k0+1), lanes 16-31 K=k0+2 / k0+3
    v2f aA, aB;
    aA.x = tile[o0 + m0 + lo];  aA.y = tile[o1 + m0 + lo];
    aB.x = tile[o0 + m1 + lo];  aB.y = tile[o1 + m1 + lo];
    // B 4x16 f32: same half-lane K split, N = lo (+16/+32 per N-tile)
    const float* Bp = Km + (size_t)ka * COUT + lo;
    v2f b0, b1, b2;
    b0.x = Bp[0];       b0.y = Bp[COUT];
    b1.x = Bp[16];      b1.y = Bp[COUT + 16];
    b2.x = Bp[32];      b2.y = Bp[COUT + 32];
    accA0 = __builtin_amdgcn_wmma_f32_16x16x4_f32(false, aA, false, b0, (short)0, accA0, false, false);
    accB0 = __builtin_amdgcn_wmma_f32_16x16x4_f32(false, aB, false, b0, (short)0, accB0, false, false);
    accA1 = __builtin_amdgcn_wmma_f32_16x16x4_f32(false, aA, false, b1, (short)0, accA1, false, false);
    accB1 = __builtin_amdgcn_wmma_f32_16x16x4_f32(false, aB, false, b1, (short)0, accB1, false, false);
    accA2 = __builtin_amdgcn_wmma_f32_16x16x4_f32(false, aA, false, b2, (short)0, accA2, false, false);
    accB2 = __builtin_amdgcn_wmma_f32_16x16x4_f32(false, aB, false, b2, (short)0, accB2, false, false);
  }

  // C/D layout: VGPR j holds (M = j + 8*hi, N = lo) -> 8 consecutive w voxels
  const size_t rowBase = (size_t)((d * S + h) * S);
  const size_t vbase0 = rowBase + (size_t)(m0 + 8 * hi);
  const size_t vbase1 = rowBase + (size_t)(m1 + 8 * hi);

  auto store8 = [&](const v8f& a, int nt, size_t vb) {
    float* o = out + (size_t)(nt * 16 + lo) * S3 + vb;
    float4 p0 = {a[0], a[1], a[2], a[3]};
    float4 p1 = {a[4], a[5], a[6], a[7]};
    *(float4*)o = p0;
    *(float4*)(o + 4) = p1;
  };
  store8(accA0, 0, vbase0); store8(accA1, 1, vbase0); store8(accA2, 2, vbase0);
  store8(accB0, 0, vbase1); store8(accB1, 1, vbase1); store8(accB2, 2, vbase1);

  // BN statistics: reduce this block's 64 voxels into shared, then global atomics
  float s0 = 0.f, q0 = 0.f, s1 = 0.f, q1 = 0.f, s2 = 0.f, q2 = 0.f;
#pragma unroll
  for (int j = 0; j < 8; ++j) {
    s0 += accA0[j] + accB0[j]; q0 += accA0[j] * accA0[j] + accB0[j] * accB0[j];
    s1 += accA1[j] + accB1[j]; q1 += accA1[j] * accA1[j] + accB1[j] * accB1[j];
    s2 += accA2[j] + accB2[j]; q2 += accA2[j] * accA2[j] + accB2[j] * accB2[j];
  }
  atomicAdd(&sSum[lo], s0);      atomicAdd(&sSq[lo], q0);
  atomicAdd(&sSum[16 + lo], s1); atomicAdd(&sSq[16 + lo], q1);
  atomicAdd(&sSum[32 + lo], s2); atomicAdd(&sSq[32 + lo], q2);
  __syncthreads();

  if (tid < 24) {
    atomicAdd(&stats[tid], sSum[tid]);
    atomicAdd(&stats[24 + tid], sSq[tid]);
  } else if (tid < 32) {
    int u = tid - 24;
    float q = sSq[24 + u * 3] + sSq[24 + u * 3 + 1] + sSq[24 + u * 3 + 2];
    atomicAdd(&stats[COUT + u], q);
  }
}

// ---------------------------------------------------------------------------
// Finalize BN stats: fin[0..23]=scale, fin[24..47]=bias', fin[48..55]=vec factor
// ---------------------------------------------------------------------------
__global__ void finalize_kernel(const float* __restrict__ stats,
                                const float* __restrict__ bn_ws,
                                const float* __restrict__ bn_bs,
                                const float* __restrict__ bn_wv,
                                float* __restrict__ fin) {
  int c = threadIdx.x;
  const float invN = 1.f / (float)S3;
  if (c < 24) {
    float mu = stats[c] * invN;
    float var = stats[24 + c] * invN - mu * mu;
    float sc = bn_ws[c] * rsqrtf(var + 1e-5f);
    fin[c] = sc;
    fin[24 + c] = bn_bs[c] - mu * sc;
  } else if (c < 32) {
    int u = c - 24;
    float n2 = stats[COUT + u] * invN;
    fin[COUT + u] = bn_wv[u] * rsqrtf(n2 + 1e-5f);
  }
}

// ---------------------------------------------------------------------------
// BN-apply + gate: 48 channels in -> 40 channels out (relu(16) | gated vec(24))
// ---------------------------------------------------------------------------
__global__ void act_kernel(const float* __restrict__ pre, const float* __restrict__ fin,
                           float* __restrict__ outY) {
  int v = blockIdx.x * blockDim.x + threadIdx.x;
  if (v >= S3) return;
  float s[24];
#pragma unroll
  for (int c = 0; c < 24; ++c) s[c] = pre[(size_t)c * S3 + v] * fin[c] + fin[24 + c];
#pragma unroll
  for (int c = 0; c < 16; ++c) outY[(size_t)c * S3 + v] = fmaxf(s[c], 0.f);
#pragma unroll
  for (int u = 0; u < 8; ++u) {
    float g = 1.f / (1.f + expf(-s[16 + u]));
    float f = fin[COUT + u] * g;
#pragma unroll
    for (int j = 0; j < 3; ++j) {
      float vv = pre[(size_t)(24 + u * 3 + j) * S3 + v];
      outY[(size_t)(16 + u * 3 + j) * S3 + v] = vv * f;
    }
  }
}

// ---------------------------------------------------------------------------
// Launch
// ---------------------------------------------------------------------------
extern "C" void kernel_launch(void* const* d_in, const int* in_sizes, int n_in,
                              void* d_out, int out_size, void* d_ws, size_t ws_size,
                              hipStream_t stream) {
  const float* x = (const float*)d_in[0];
  const float* w1 = (const float*)d_in[1];
  const float* sc1 = (const float*)d_in[2];
  const float* w2 = (const float*)d_in[3];
  const float* sc2_s = (const float*)d_in[4];
  const float* sc2_v = (const float*)d_in[5];
  const float* bn1_ws = (const float*)d_in[6];
  const float* bn1_bs = (const float*)d_in[7];
  const float* bn1_wv = (const float*)d_in[8];
  const float* bn2_ws = (const float*)d_in[9];
  const float* bn2_bs = (const float*)d_in[10];
  const float* bn2_wv = (const float*)d_in[11];

  float* ws = (float*)d_ws;
  float* Km1 = ws;                 // 27*8*48  = 10368
  float* Km2 = ws + 10368;         // 27*40*48 = 51840
  float* stats1 = ws + 62208;      // 56
  float* stats2 = ws + 62264;      // 56
  float* fin1 = ws + 62320;        // 56
  float* fin2 = ws + 62376;        // 56
  float* pre = ws + 65536;         // 48 * 64^3 (reused for both convs)
  float* ybuf = pre + (size_t)COUT * S3;  // 40 * 64^3
  float* zout = (float*)d_out;     // 40 * 64^3

  // smem: tile + 96 stats floats + koff ints
  const size_t sm1 = (size_t)(8 * 9 * ROWLEN + 2 * COUT) * 4 + (size_t)(8 * 27) * 4;
  const size_t sm2 = (size_t)(40 * 9 * ROWLEN + 2 * COUT) * 4 + (size_t)(40 * 27) * 4;
  // conv2 tile exceeds the 64KB default dynamic-LDS cap; opt in (320KB/WG on CDNA5)
  (void)hipFuncSetAttribute((const void*)conv_wmma_kernel,
                            hipFuncAttributeMaxDynamicSharedMemorySize, (int)sm2);

  // 1) build conv kernels (skips folded in) + zero stats region
  prep_kernel<<<64, 256, 0, stream>>>(w1, sc1, w2, sc2_s, sc2_v, Km1, Km2, stats1);

  // 2) conv1: cin=8, KT=216
  conv_wmma_kernel<<<S * S, 64, sm1, stream>>>(x, Km1, pre, stats1, 8);
  finalize_kernel<<<1, 32, 0, stream>>>(stats1, bn1_ws, bn1_bs, bn1_wv, fin1);
  act_kernel<<<(S3 + 255) / 256, 256, 0, stream>>>(pre, fin1, ybuf);

  // 3) conv2: cin=40, KT=1080
  conv_wmma_kernel<<<S * S, 64, sm2, stream>>>(ybuf, Km2, pre, stats2, 40);
  finalize_kernel<<<1, 32, 0, stream>>>(stats2, bn2_ws, bn2_bs, bn2_wv, fin2);
  act_kernel<<<(S3 + 255) / 256, 256, 0, stream>>>(pre, fin2, zout);
}